// GammaLambdaLearner_77412490543767
// MI455X (gfx1250) — compile-verified
//
#include <hip/hip_runtime.h>
#include <hip/hip_bf16.h>
#include <stdint.h>

// Problem sizes (fixed by the reference)
#define B_DIM 32768
#define S_DIM 512
#define EPSF  1e-8f

// Tiling: 64 rows per block (2 wave32 waves), each wave owns 32 rows.
// Time processed in chunks of 32 steps, double-buffered in LDS via TDM.
#define NB   64            // batch rows per block == blockDim.x
#define WR   32            // rows per wave (wave32: one row per lane)
#define CT   32            // timesteps per chunk (32 floats = 128B row segment)
#define NC   (S_DIM / CT)  // 16 chunks
#define ROWW (CT + 1)      // padded LDS row stride in dwords (33 -> conflict-free)

typedef unsigned int v4u __attribute__((ext_vector_type(4)));
typedef int          v4i __attribute__((ext_vector_type(4)));
typedef int          v8i __attribute__((ext_vector_type(8)));

// Low 32 bits of a generic pointer to __shared__ == byte offset in LDS.
__device__ __forceinline__ uint32_t lds_off_of(const void* p) {
  return (uint32_t)(uintptr_t)p;
}

// Issue one 2D TDM tile load: WR rows x CT elements of f32, row stride
// `row_stride_elems`, into LDS at lds_byte_off with 1-dword padding per
// 32-dword row (D# pad_enable, pad_interval=32dw, pad_amount=1dw).
__device__ __forceinline__ void tdm_load_tile(uint32_t lds_byte_off,
                                              const float* gbase,
                                              uint32_t elem_off,
                                              uint32_t row_stride_elems) {
  uint64_t ga = (uint64_t)(uintptr_t)gbase + 4ull * (uint64_t)elem_off;

  // D# group 0: count=1, lds_addr, global_addr[56:0], type=2 ("image")
  v4u g0;
  g0[0] = 1u;
  g0[1] = lds_byte_off;
  g0[2] = (uint32_t)ga;
  g0[3] = (uint32_t)((ga >> 32) & 0x01FFFFFFu) | (2u << 30);

  // D# group 1:
  //  word0: workgroup_mask=0 | data_size=2 (4B)<<16 | pad_enable<<20 |
  //         pad_interval=4 (32 dwords)<<22 | pad_amount=0 (1 dword)<<25
  //  word1: atomic_barrier_addr=0 | tensor_dim0[15:0]<<16
  //  word2: tensor_dim0[31:16]    | tensor_dim1[15:0]<<16
  //  word3: tensor_dim1[31:16]    | tile_dim0<<16
  //  word4: tile_dim1             | tile_dim2=0
  //  word5: tensor_dim0_stride[31:0]
  //  word6: tensor_dim0_stride[47:32] | tensor_dim1_stride[15:0]<<16 (=0)
  //  word7: tensor_dim1_stride[47:16] (=0)
  v8i g1;
  g1[0] = (int)((2u << 16) | (1u << 20) | (4u << 22));
  g1[1] = (int)(((uint32_t)CT) << 16);   // tensor_dim0 = CT
  g1[2] = (int)(((uint32_t)WR) << 16);   // tensor_dim1 = WR
  g1[3] = (int)(((uint32_t)CT) << 16);   // tile_dim0   = CT
  g1[4] = (int)WR;                       // tile_dim1   = WR, tile_dim2 = 0
  g1[5] = (int)row_stride_elems;         // tensor_dim0_stride
  g1[6] = 0;
  g1[7] = 0;

  v4i z4 = {0, 0, 0, 0};                 // groups 2/3: dims 2..4 unused
#if __has_include(<hip/amd_detail/amd_gfx1250_TDM.h>)
  v8i z8 = {0, 0, 0, 0, 0, 0, 0, 0};
  __builtin_amdgcn_tensor_load_to_lds(g0, g1, z4, z4, z8, 0);
#else
  __builtin_amdgcn_tensor_load_to_lds(g0, g1, z4, z4, 0);
#endif
}

__global__ __launch_bounds__(NB)
void GammaLambdaLearner_77412490543767_kernel(const float* __restrict__ values,
                                              const float* __restrict__ rewards,
                                              const float* __restrict__ dones,
                                              const float* __restrict__ raw_gamma,
                                              const float* __restrict__ raw_lambd,
                                              float* __restrict__ out) {
  // Double-buffered TDM destination tiles (padded rows, bank-conflict free).
  __shared__ float tR[2][NB * ROWW];
  __shared__ float tD[2][NB * ROWW];
  __shared__ float tV[2][NB * ROWW];
  __shared__ float lam_s[S_DIM];

  const int tid = (int)threadIdx.x;                       // 0..63, one batch row
  const int wv  = __builtin_amdgcn_readfirstlane(tid >> 5); // wave id (uniform)
  const uint32_t rowBase = (uint32_t)blockIdx.x * NB + (uint32_t)wv * WR;
  const uint32_t ldsW    = (uint32_t)(wv * WR * ROWW * 4); // wave's LDS slice

  // Prologue: kick off DMA for the last time-chunk (scan is reverse) -> buf 0.
  {
    const uint32_t t0 = (uint32_t)(NC - 1) * CT;
    tdm_load_tile(lds_off_of(&tR[0][0]) + ldsW, rewards, rowBase * S_DIM + t0, S_DIM);
    tdm_load_tile(lds_off_of(&tD[0][0]) + ldsW, dones,   rowBase * S_DIM + t0, S_DIM);
    tdm_load_tile(lds_off_of(&tV[0][0]) + ldsW, values,
                  rowBase * (S_DIM + 1) + t0 + 1, S_DIM + 1);
  }

  // Stage lambdas (shared across the block) while the DMA flies.
  for (int i = tid; i < S_DIM; i += NB)
    lam_s[i] = fmaxf(tanhf(raw_lambd[i]), EPSF);

  const float g = fmaxf(tanhf(raw_gamma[0]), EPSF);
  const uint32_t b = (uint32_t)blockIdx.x * NB + (uint32_t)tid;
  float ret = values[(size_t)b * (S_DIM + 1) + S_DIM];   // init = values[:, -1]
  float* orow = out + (size_t)b * S_DIM;

  __syncthreads();                                       // lam_s visible
  asm volatile("" ::: "memory");
  __builtin_amdgcn_s_wait_tensorcnt(0);                  // wave's tiles landed
  asm volatile("" ::: "memory");

  int cur = 0;
  for (int c = NC - 1; c >= 0; --c) {
    // Issue next (earlier) chunk into the other buffer; each wave only ever
    // touches its own rows, so no cross-wave barrier is needed.
    if (c > 0) {
      const uint32_t t0n = (uint32_t)(c - 1) * CT;
      const int nxt = cur ^ 1;
      tdm_load_tile(lds_off_of(&tR[nxt][0]) + ldsW, rewards, rowBase * S_DIM + t0n, S_DIM);
      tdm_load_tile(lds_off_of(&tD[nxt][0]) + ldsW, dones,   rowBase * S_DIM + t0n, S_DIM);
      tdm_load_tile(lds_off_of(&tV[nxt][0]) + ldsW, values,
                    rowBase * (S_DIM + 1) + t0n + 1, S_DIM + 1);
    }

    const int t0 = c * CT;
    const float* rR = &tR[cur][tid * ROWW];
    const float* rD = &tD[cur][tid * ROWW];
    const float* rV = &tV[cur][tid * ROWW];

#pragma unroll
    for (int tt = CT - 4; tt >= 0; tt -= 4) {
      float4 o;
      {
        const float lam = lam_s[t0 + tt + 3];
        const float boot = fmaf(lam, ret, (1.0f - lam) * rV[tt + 3]);
        ret = fmaf(g * (1.0f - rD[tt + 3]), boot, rR[tt + 3]);
        o.w = ret;
      }
      {
        const float lam = lam_s[t0 + tt + 2];
        const float boot = fmaf(lam, ret, (1.0f - lam) * rV[tt + 2]);
        ret = fmaf(g * (1.0f - rD[tt + 2]), boot, rR[tt + 2]);
        o.z = ret;
      }
      {
        const float lam = lam_s[t0 + tt + 1];
        const float boot = fmaf(lam, ret, (1.0f - lam) * rV[tt + 1]);
        ret = fmaf(g * (1.0f - rD[tt + 1]), boot, rR[tt + 1]);
        o.y = ret;
      }
      {
        const float lam = lam_s[t0 + tt];
        const float boot = fmaf(lam, ret, (1.0f - lam) * rV[tt]);
        ret = fmaf(g * (1.0f - rD[tt]), boot, rR[tt]);
        o.x = ret;
      }
      *reinterpret_cast<float4*>(orow + t0 + tt) = o;    // 16B-aligned store
    }

    if (c > 0) {
      asm volatile("" ::: "memory");
      __builtin_amdgcn_s_wait_tensorcnt(0);              // next buffer ready
      asm volatile("" ::: "memory");
    }
    cur ^= 1;
  }
}

extern "C" void kernel_launch(void* const* d_in, const int* in_sizes, int n_in,
                              void* d_out, int out_size, void* d_ws, size_t ws_size,
                              hipStream_t stream) {
  (void)in_sizes; (void)n_in; (void)out_size; (void)d_ws; (void)ws_size;
  const float* values    = (const float*)d_in[0];   // (B, S+1, 1)
  const float* rewards   = (const float*)d_in[1];   // (B, S, 1)
  const float* dones     = (const float*)d_in[2];   // (B, S, 1)
  const float* raw_gamma = (const float*)d_in[3];   // (1,)
  const float* raw_lambd = (const float*)d_in[4];   // (S,)
  float* out = (float*)d_out;                       // (B, S, 1)

  dim3 grid(B_DIM / NB);   // 512 blocks
  dim3 block(NB);          // 64 threads = 2 wave32 waves
  hipLaunchKernelGGL(GammaLambdaLearner_77412490543767_kernel,
                     grid, block, 0, stream,
                     values, rewards, dones, raw_gamma, raw_lambd, out);
}